// MoELoRALayer_46334107189262
// MI455X (gfx1250) — compile-verified
//
#include <hip/hip_runtime.h>
#include <hip/hip_bf16.h>
#include <math.h>

#define NTOK 8192
#define DDIM 1024
#define NEXP 8
#define RNK  16
#define ODIM 3072
#define XPAD 1026   // x tile row stride (floats): even (8B align) + bank spread
#define WTP  1032   // transposed-W row stride (floats): even + bank spread
#define NEPS 0.01f

typedef __attribute__((ext_vector_type(2))) float v2f;
typedef __attribute__((ext_vector_type(8))) float v8f;

static __device__ __forceinline__ v8f wmma4(v2f a, v2f b, v8f c) {
  // V_WMMA_F32_16X16X4_F32 : D = A(16x4) * B(4x16) + C
  return __builtin_amdgcn_wmma_f32_16x16x4_f32(
      /*neg_a=*/false, a, /*neg_b=*/false, b,
      /*c_mod=*/(short)0, c, /*reuse_a=*/false, /*reuse_b=*/false);
}

// Async 16B-per-lane copy global -> LDS (GLOBAL_LOAD_ASYNC_TO_LDS_B128, ASYNCcnt).
static __device__ __forceinline__ void async_copy_b128(unsigned lds_off,
                                                       unsigned long long gaddr) {
  asm volatile("global_load_async_to_lds_b128 %0, %1, off"
               :: "v"(lds_off), "v"(gaddr) : "memory");
}
static __device__ __forceinline__ void wait_async() {
  asm volatile("s_wait_asynccnt 0x0" ::: "memory");
}

__global__ void moe_init_kernel(int* __restrict__ counts, float* __restrict__ stats) {
  int t = threadIdx.x;
  if (t < NEXP) counts[t] = 0;
  if (t < 2 * NEXP) stats[t] = 0.f;   // importance[8] + load[8]
}

// ---------------- Kernel 1: gating + routing ----------------
// One block = 16 tokens. [clean | softplus-arg] = x_tile @ [w_gate|w_noise]
// via f32 WMMA, K split over 8 waves. x tile staged by async-DMA while the
// W transpose staging runs; inner loop = 2x ds_load_b64 + 1 wmma.
__global__ __launch_bounds__(256)
void moe_gating_kernel(const float* __restrict__ x,
                       const float* __restrict__ wg,
                       const float* __restrict__ wn,
                       const float* __restrict__ noise,
                       float* __restrict__ imp,
                       float* __restrict__ loadv,
                       int* __restrict__ counts,
                       int* __restrict__ lists) {
  __shared__ float xs[16 * XPAD];    // ~64KB x tile
  __shared__ float wt[16 * WTP];     // ~66KB transposed [w_gate|w_noise]
  __shared__ float part[8 * 256];    // per-wave partial C tiles
  __shared__ float ltile[256];       // reduced 16x16 logits tile
  __shared__ float accI[NEXP];
  __shared__ float accL[NEXP];

  const int tid = threadIdx.x;
  const int n0  = blockIdx.x * 16;
  if (tid < NEXP) { accI[tid] = 0.f; accL[tid] = 0.f; }

  // 1) kick off async x-tile staging (16B per lane per row)
  {
    const unsigned lds0 = (unsigned)(size_t)(void*)&xs[0];
    #pragma unroll 4
    for (int r = 0; r < 16; ++r) {
      async_copy_b128(lds0 + (unsigned)((r * XPAD + tid * 4) * 4),
                      (unsigned long long)(size_t)&x[(size_t)(n0 + r) * DDIM + tid * 4]);
    }
  }
  // 2) overlap: stage W transposed: wt[n][k], n<8 -> w_gate, n>=8 -> w_noise
  {
    const int n = tid & 15;
    const float* src = ((n < 8) ? wg : wn) + (n & 7);
    const int kb = tid >> 4;                   // 0..15
    #pragma unroll 8
    for (int p = 0; p < 64; ++p) {
      const int k = p * 16 + kb;
      wt[n * WTP + k] = src[k * NEXP];
    }
  }
  wait_async();
  __syncthreads();

  const int wave = tid >> 5, lane = tid & 31;
  const int m = lane & 15, hi = lane >> 4;

  v8f acc = {0.f, 0.f, 0.f, 0.f, 0.f, 0.f, 0.f, 0.f};
  {
    const float* ap = &xs[m * XPAD + wave * 128 + hi * 2];
    const float* bp = &wt[m * WTP + wave * 128 + hi * 2];
    #pragma unroll 8
    for (int it = 0; it < 32; ++it) {
      v2f a = *(const v2f*)(ap);   // x[m][kk, kk+1]
      v2f b = *(const v2f*)(bp);   // W[kk, kk+1][n=m]
      acc = wmma4(a, b, acc);
      ap += 4; bp += 4;
    }
  }
  #pragma unroll
  for (int v = 0; v < 8; ++v) part[wave * 256 + (v + hi * 8) * 16 + m] = acc[v];
  __syncthreads();

  { // K-split reduction
    float s = 0.f;
    #pragma unroll
    for (int w = 0; w < 8; ++w) s += part[w * 256 + tid];
    ltile[tid] = s;
  }
  __syncthreads();

  if (tid < 16) {
    const int n = n0 + tid;
    float clean[NEXP], stdv[NEXP], lg[NEXP];
    #pragma unroll
    for (int e = 0; e < NEXP; ++e) {
      clean[e] = ltile[tid * 16 + e];
      float nr = ltile[tid * 16 + 8 + e];
      float sp = (nr > 20.f) ? nr : log1pf(expf(nr));   // softplus
      stdv[e] = sp + NEPS;
      lg[e] = clean[e] + noise[n * NEXP + e] * stdv[e];
    }
    float best = -INFINITY, sec = -INFINITY; int be = 0;
    #pragma unroll
    for (int e = 0; e < NEXP; ++e) {
      float v = lg[e];
      if (v > best) { sec = best; best = v; be = e; }
      else if (v > sec) sec = v;
    }
    #pragma unroll
    for (int e = 0; e < NEXP; ++e) {
      float thr = (lg[e] > sec) ? sec : best;            // thr_in / thr_out
      float z = (clean[e] - thr) / stdv[e];
      float p = 0.5f * erfcf(-z * 0.70710678118654752f); // normcdf
      atomicAdd(&accL[e], p);
    }
    atomicAdd(&accI[be], 1.0f);                          // K=1 softmax gate == 1
    int pos = atomicAdd(&counts[be], 1);
    lists[be * NTOK + pos] = n;
  }
  __syncthreads();
  if (tid < NEXP) atomicAdd(&imp[tid], accI[tid]);
  else if (tid < 2 * NEXP) atomicAdd(&loadv[tid - NEXP], accL[tid - NEXP]);
}

// ---------------- Kernel 2: grouped LoRA ----------------
// Block = 16 tokens of ONE expert: h = x_tile @ lora_a[e]^T (WMMA, K-split),
// then out_tile(16x3072) = h @ lora_b[e]^T (192 N-tiles x 4 WMMAs).
__global__ __launch_bounds__(256)
void moe_lora_kernel(const float* __restrict__ x,
                     const float* __restrict__ la,   // (E,R,D)
                     const float* __restrict__ lb,   // (E,3D,R)
                     const int* __restrict__ counts,
                     const int* __restrict__ lists,
                     float* __restrict__ out) {
  __shared__ float xs[16 * XPAD];
  __shared__ float part[8 * 256];
  __shared__ float ht[16 * 18];      // h tile, padded
  __shared__ int   stile[16];

  const int e = blockIdx.y;
  const int cnt = counts[e];
  const int base = blockIdx.x * 16;
  if (base >= cnt) return;           // uniform exit

  const int tid = threadIdx.x;
  if (tid < 16) {
    int idx = base + tid; if (idx > cnt - 1) idx = cnt - 1;  // clamp tail
    stile[tid] = lists[e * NTOK + idx];
  }
  __syncthreads();
  // async x-tile staging (gathered rows)
  {
    const unsigned lds0 = (unsigned)(size_t)(void*)&xs[0];
    #pragma unroll 4
    for (int r = 0; r < 16; ++r) {
      async_copy_b128(lds0 + (unsigned)((r * XPAD + tid * 4) * 4),
                      (unsigned long long)(size_t)&x[(size_t)stile[r] * DDIM + tid * 4]);
    }
  }
  wait_async();
  __syncthreads();

  const int wave = tid >> 5, lane = tid & 31;
  const int m = lane & 15, hi = lane >> 4;

  // h = x_tile @ lora_a[e]^T   (16x16, K=1024 split over 8 waves)
  v8f acc = {0.f, 0.f, 0.f, 0.f, 0.f, 0.f, 0.f, 0.f};
  {
    const float* ap = &xs[m * XPAD + wave * 128 + hi * 2];
    const float* bp = la + (size_t)e * RNK * DDIM + m * DDIM + wave * 128 + hi * 2;
    #pragma unroll 8
    for (int it = 0; it < 32; ++it) {
      v2f a = *(const v2f*)(ap);     // x[m][kk, kk+1]
      v2f b = *(const v2f*)(bp);     // lora_a[e][r=m][kk, kk+1]
      acc = wmma4(a, b, acc);
      ap += 4; bp += 4;
    }
  }
  #pragma unroll
  for (int v = 0; v < 8; ++v) part[wave * 256 + (v + hi * 8) * 16 + m] = acc[v];
  __syncthreads();
  {
    float s = 0.f;
    #pragma unroll
    for (int w = 0; w < 8; ++w) s += part[w * 256 + tid];
    ht[(tid >> 4) * 18 + (tid & 15)] = s;
  }
  __syncthreads();

  // hoist A fragments (same for every N-tile) and store masks/rows
  v2f af[4];
  #pragma unroll
  for (int j = 0; j < 4; ++j) af[j] = *(const v2f*)&ht[m * 18 + j * 4 + hi * 2];
  int tks[8]; bool oks[8];
  #pragma unroll
  for (int v = 0; v < 8; ++v) {
    const int row = v + hi * 8;
    oks[v] = (base + row) < cnt;
    tks[v] = stile[row];
  }

  // out tile: per-lane B base, tiles advance by 256 floats
  const float* bbase = lb + (size_t)e * ODIM * RNK + m * RNK + hi * 2;
  for (int ot = wave; ot < ODIM / 16; ot += 8) {
    const int ob = ot * 16;
    const float* bp = bbase + (size_t)ob * RNK;
    v8f c = {0.f, 0.f, 0.f, 0.f, 0.f, 0.f, 0.f, 0.f};
    #pragma unroll
    for (int j = 0; j < 4; ++j) {
      v2f b = *(const v2f*)(bp + j * 4);   // B[k][o] = lora_b[e][o][k]
      c = wmma4(af[j], b, c);
    }
    #pragma unroll
    for (int v = 0; v < 8; ++v) {
      if (oks[v]) out[(size_t)tks[v] * ODIM + ob + m] = c[v];
    }
  }
}

extern "C" void kernel_launch(void* const* d_in, const int* in_sizes, int n_in,
                              void* d_out, int out_size, void* d_ws, size_t ws_size,
                              hipStream_t stream) {
  const float* x     = (const float*)d_in[0];
  const float* wg    = (const float*)d_in[1];
  const float* wn    = (const float*)d_in[2];
  const float* la    = (const float*)d_in[3];
  const float* lb    = (const float*)d_in[4];
  const float* noise = (const float*)d_in[5];

  float* out   = (float*)d_out;
  float* imp   = out + (size_t)NTOK * ODIM;   // importance[8]
  float* loadv = imp + NEXP;                  // load[8]

  int* counts = (int*)d_ws;                   // 8 counters
  int* lists  = counts + 64;                  // 256B offset: E x N token lists

  hipLaunchKernelGGL(moe_init_kernel, dim3(1), dim3(32), 0, stream, counts, imp);
  hipLaunchKernelGGL(moe_gating_kernel, dim3(NTOK / 16), dim3(256), 0, stream,
                     x, wg, wn, noise, imp, loadv, counts, lists);
  hipLaunchKernelGGL(moe_lora_kernel, dim3(NTOK / 16, NEXP), dim3(256), 0, stream,
                     x, la, lb, counts, lists, out);
}